// PRCNv1_22703197126947
// MI455X (gfx1250) — compile-verified
//
#include <hip/hip_runtime.h>
#include <hip/hip_bf16.h>
#include <math.h>

typedef __attribute__((ext_vector_type(2))) float v2f;
typedef __attribute__((ext_vector_type(8))) float v8f;

// Problem constants (from reference)
#define C_IN     64
#define OUTC_    64
#define CONV_CH  4096      // OUTC * C_IN
#define G_       8
#define AVG_     32        // C_IN*EXP//G
#define NB       8
#define H_       28
#define W_       28
#define HW_      784       // 28*28 = 49 tiles of 16 pixels
#define NT_      49
#define OUT_ELEMS (NB * 64 * HW_)   // 401408 floats for `out`

// ---------------------------------------------------------------------------
// Kernel 1: grouped 3x3 conv as per-channel GEMM on the matrix pipe.
//   For fixed input channel c: D[64 x 784] = Wc[64 x 9] * patch[9 x 784] + b
//   grid = (nt=49, c-chunk=8, b=8), block = 256 = 8 waves; wave w owns
//   channel c = blockIdx.y*8 + w. No integer division in the decomposition.
//   3x V_WMMA_F32_16X16X4_F32 per 16x16 output tile (K=9 padded to 12),
//   4 M-tiles per wave reusing the B (patch) operands.
// ---------------------------------------------------------------------------
__global__ __launch_bounds__(256)
void prcn_conv_wmma(const float* __restrict__ x,
                    const float* __restrict__ Wc,
                    const float* __restrict__ bias,
                    float* __restrict__ out_conv)
{
    const int nt = blockIdx.x;                          // 0..48
    const int c  = blockIdx.y * 8 + (threadIdx.x >> 5); // 0..63
    const int b  = blockIdx.z;                          // 0..7

    const int lane = threadIdx.x & 31;
    const int half = lane >> 4;     // 0: lanes 0-15, 1: lanes 16-31
    const int l16  = lane & 15;

    // This lane's pixel (both half-waves cover the same 16 pixels; they
    // supply different K rows of the B matrix).
    const int p = nt * 16 + l16;
    const int h = p / W_;           // constant divisor -> single mul_hi
    const int w = p % W_;

    // Load the 9 taps of the 3x3 patch, zero-padded at borders.
    // Branch-free (selects), so EXEC is all-1s at the WMMAs.
    float t[9];
    const float* xc = x + ((size_t)(b * C_IN + c)) * HW_;
#pragma unroll
    for (int dy = -1; dy <= 1; ++dy) {
#pragma unroll
        for (int dx = -1; dx <= 1; ++dx) {
            const int hh = h + dy, ww = w + dx;
            const bool ok = (hh >= 0) & (hh < H_) & (ww >= 0) & (ww < W_);
            t[(dy + 1) * 3 + (dx + 1)] = ok ? xc[hh * W_ + ww] : 0.0f;
        }
    }

    // B-matrix chunks (4x16 f32 each): lane half selects K rows.
    //   chunk0: K0..K3, chunk1: K4..K7, chunk2: K8 + zero pad.
    v2f bk0, bk1, bk2;
    bk0.x = t[0 + 2 * half]; bk0.y = t[1 + 2 * half];
    bk1.x = t[4 + 2 * half]; bk1.y = t[5 + 2 * half];
    bk2.x = half ? 0.0f : t[8];
    bk2.y = 0.0f;

    const float* wbase = Wc + ((size_t)(c * OUTC_)) * 9; // Wc[(64c+m)*9 + k]
    const float* bbase = bias + c * OUTC_;

#pragma unroll
    for (int mt = 0; mt < 4; ++mt) {
        // A-matrix chunks (16x4 f32): lane l16 is the filter row M.
        const float* wm = wbase + (mt * 16 + l16) * 9;
        v2f ak0, ak1, ak2;
        ak0.x = wm[0 + 2 * half]; ak0.y = wm[1 + 2 * half];
        ak1.x = wm[4 + 2 * half]; ak1.y = wm[5 + 2 * half];
        ak2.x = half ? 0.0f : wm[8];
        ak2.y = 0.0f;

        // Initialize accumulator with the bias: C/D VGPR r holds output
        // channel (mt*16 + r + 8*half) for this lane.
        v8f acc;
#pragma unroll
        for (int r = 0; r < 8; ++r)
            acc[r] = bbase[mt * 16 + r + 8 * half];

        // D = A*B + C, chained over the K dimension (9 -> 3x K=4).
        acc = __builtin_amdgcn_wmma_f32_16x16x4_f32(false, ak0, false, bk0,
                                                    (short)0, acc, false, false);
        acc = __builtin_amdgcn_wmma_f32_16x16x4_f32(false, ak1, false, bk1,
                                                    (short)0, acc, false, false);
        acc = __builtin_amdgcn_wmma_f32_16x16x4_f32(false, ak2, false, bk2,
                                                    (short)0, acc, false, false);

        // Store 8 rows; each half-wave writes 16 contiguous pixels per row.
        const size_t base =
            ((size_t)(b * CONV_CH + c * OUTC_ + mt * 16 + 8 * half)) * HW_
            + (size_t)(nt * 16 + l16);
#pragma unroll
        for (int r = 0; r < 8; ++r)
            out_conv[base + (size_t)r * HW_] = acc[r];
    }
}

// ---------------------------------------------------------------------------
// Kernel 2: gather (index % 4096) -> max over 8 -> mean over 32.
// out_conv (103 MB) fits in the 192 MB L2, so the 4x logical re-read stays
// on-chip. index reads are block-uniform (oc = blockIdx.y) -> scalar loads.
// ---------------------------------------------------------------------------
__global__ __launch_bounds__(256)
void prcn_gather_max_mean(const float* __restrict__ out_conv,
                          const int* __restrict__ index,
                          float* __restrict__ out)
{
    const int p  = blockIdx.x * 256 + threadIdx.x;  // pixel
    const int oc = blockIdx.y;                      // 0..63
    const int b  = blockIdx.z;                      // 0..7
    if (p >= HW_) return;

    const float* base = out_conv + (size_t)b * CONV_CH * HW_ + p;
    const int*   idx  = index + oc * (AVG_ * G_);   // 256 uniform ints

    float sum = 0.0f;
#pragma unroll 1
    for (int a = 0; a < AVG_; ++a) {
        float m = -INFINITY;
#pragma unroll
        for (int g = 0; g < G_; ++g) {
            const int ch = idx[a * G_ + g] & (CONV_CH - 1); // % 4096
            m = fmaxf(m, base[(size_t)ch * HW_]);
        }
        sum += m;
    }
    out[((size_t)(b * 64 + oc)) * HW_ + p] = sum * (1.0f / AVG_);
}

// ---------------------------------------------------------------------------
// Host entry. Inputs: x, Wc, b, index. Output: [out | out_conv] concatenated.
// ---------------------------------------------------------------------------
extern "C" void kernel_launch(void* const* d_in, const int* in_sizes, int n_in,
                              void* d_out, int out_size, void* d_ws, size_t ws_size,
                              hipStream_t stream)
{
    (void)in_sizes; (void)n_in; (void)out_size; (void)d_ws; (void)ws_size;

    const float* x    = (const float*)d_in[0];
    const float* Wc   = (const float*)d_in[1];
    const float* bias = (const float*)d_in[2];
    const int*   idx  = (const int*)d_in[3];

    float* out      = (float*)d_out;             // 8*64*784 floats
    float* out_conv = (float*)d_out + OUT_ELEMS; // 8*4096*784 floats

    // Kernel 1: grid (49 n-tiles, 8 c-chunks, 8 b); 8 waves per block.
    prcn_conv_wmma<<<dim3(NT_, 8, NB), dim3(256), 0, stream>>>(x, Wc, bias, out_conv);

    // Kernel 2: grid (pixel chunks, oc, b); same stream -> ordered after k1.
    dim3 grid2((HW_ + 255) / 256, 64, NB);
    prcn_gather_max_mean<<<grid2, dim3(256), 0, stream>>>(out_conv, idx, out);
}